// PG_STGNN_83717502534029
// MI455X (gfx1250) — compile-verified
//
#include <hip/hip_runtime.h>
#include <hip/hip_bf16.h>
#include <math.h>

// ---------------------------------------------------------------------------
// PG-STGNN on MI455X (gfx1250): all GEMM stages on V_WMMA_F32_16X16X4_F32
// (wave32, fp32 matrix pipe — matches reference fp32 numerics).
//
// VGPR layout (CDNA5 ISA 7.12.2):
//  A (16x4 f32):  lanes 0-15: row M=l, v0=K0,v1=K1 ; lanes 16-31: row M=l-16, v0=K2,v1=K3
//  B (4x16 f32):  lanes 0-15: col N=l, v0=K0,v1=K1 ; lanes 16-31: col N=l-16, v0=K2,v1=K3
//  C/D (16x16):   VGPR r, lanes 0-15 -> (M=r, N=l); lanes 16-31 -> (M=r+8, N=l-16)
// ---------------------------------------------------------------------------

typedef float v2f __attribute__((ext_vector_type(2)));
typedef float v8f __attribute__((ext_vector_type(8)));

#define BB    16
#define NVTX  512
#define TT    168
#define FIN   8
#define HIDD  64
#define NCOL  (TT * FIN)        // 1344
#define RTOT  (BB * NVTX)       // 8192

__device__ __forceinline__ v8f wmma_f32_k4(v2f a, v2f b, v8f c) {
  return __builtin_amdgcn_wmma_f32_16x16x4_f32(false, a, false, b, (short)0, c,
                                               false, false);
}

__device__ __forceinline__ float sigm(float x) { return 1.0f / (1.0f + expf(-x)); }

// ---------------------------------------------------------------------------
// Stage 1: lagged graph aggregation.
// agg[b][v][c] = sum_{lag} sum_u A_lag[v,u] * Xb[u, c - 8*lag]   (c = t*8+f)
// Each wave computes a 32x16 tile (2 accumulators sharing one B fetch);
// the lag shift is handled branch-free: clamped address + 0/1 mask multiply.
// 4 waves per block walk identical B addresses -> L0 reuse.
// ---------------------------------------------------------------------------
__global__ __launch_bounds__(128) void agg_wmma_kernel(
    const float* __restrict__ x,     // [B, N, T, F] == per-b [512, 1344]
    const float* __restrict__ A,     // [4, 512, 512]
    float* __restrict__ agg)         // [B, 512, 1344]
{
  const int wave = threadIdx.x >> 5;
  const int lane = threadIdx.x & 31;
  const int l16  = lane & 15;
  const int half = lane >> 4;                  // 0: K{0,1}, 1: K{2,3}
  const int c0 = blockIdx.x * 16;              // column tile (t*8+f)
  const int v0 = blockIdx.y * 128 + wave * 32; // vertex tile (32 rows / wave)
  const int b  = blockIdx.z;

  const float* Xb = x + (size_t)b * NVTX * NCOL;
  v8f acc0 = {0.f, 0.f, 0.f, 0.f, 0.f, 0.f, 0.f, 0.f};
  v8f acc1 = acc0;

  for (int lag = 0; lag < 4; ++lag) {
    const float* Ak0 = A + (size_t)lag * NVTX * NVTX
                         + (size_t)(v0 + l16) * NVTX + half * 2;
    const float* Ak1 = Ak0 + (size_t)16 * NVTX;
    const int   cc  = c0 + l16 - 8 * lag;       // shifted column (per lane)
    const int   ccc = cc < 0 ? 0 : cc;          // clamped, always in-bounds
    const float msk = cc < 0 ? 0.0f : 1.0f;     // zero-padding mask
    const float* Xc = Xb + ccc;
    #pragma unroll 4
    for (int kb = 0; kb < NVTX; kb += 4) {
      v2f a0 = *(const v2f*)(Ak0 + kb);
      v2f a1 = *(const v2f*)(Ak1 + kb);
      const int u0 = kb + half * 2;
      v2f bv;
      bv.x = Xc[(size_t)u0 * NCOL] * msk;       // unconditional load + mask
      bv.y = Xc[(size_t)(u0 + 1) * NCOL] * msk;
      acc0 = wmma_f32_k4(a0, bv, acc0);
      acc1 = wmma_f32_k4(a1, bv, acc1);
    }
  }

  float* Ob = agg + (size_t)b * NVTX * NCOL;
  #pragma unroll
  for (int r = 0; r < 8; ++r) {
    Ob[(size_t)(v0 + r + half * 8) * NCOL + (c0 + l16)] = acc0[r];
    Ob[(size_t)(v0 + 16 + r + half * 8) * NCOL + (c0 + l16)] = acc1[r];
  }
}

// ---------------------------------------------------------------------------
// Stage 2: 8->64 projection + exact GELU, writing g in LSTM layout [T, R, H].
// ---------------------------------------------------------------------------
__global__ __launch_bounds__(256) void proj_gelu_kernel(
    const float* __restrict__ agg,   // [R, 1344] with r = b*512+v
    const float* __restrict__ Ww,    // [64, 8]
    const float* __restrict__ Wb,    // [64]
    float* __restrict__ g)           // [T, R, 64]
{
  const size_t idx = (size_t)blockIdx.x * blockDim.x + threadIdx.x;
  if (idx >= (size_t)RTOT * TT * HIDD) return;
  const int    h  = (int)(idx & (HIDD - 1));
  const size_t rt = idx >> 6;
  const int    t  = (int)(rt % TT);
  const size_t r  = rt / TT;

  const float* av = agg + r * NCOL + (size_t)t * FIN;
  const float* w  = Ww + (size_t)h * FIN;
  float s = Wb[h];
  #pragma unroll
  for (int f = 0; f < FIN; ++f) s += av[f] * w[f];
  const float gl = 0.5f * s * (1.0f + erff(s * 0.7071067811865475f));
  g[((size_t)t * RTOT + r) * HIDD + h] = gl;
}

// ---------------------------------------------------------------------------
// Stage 3: one LSTM timestep. 4 waves/block; wave w owns hidden cols
// [16w, 16w+16) for rows [16*blockIdx.x, +16) and accumulates the i/f/g/o
// gate tiles (shared A operand, K = 64(g_t) + 64(h_prev)). Epilogue fuses
// the cell update and the H->1 output FC (deterministic LDS reduce).
// ---------------------------------------------------------------------------
__global__ __launch_bounds__(128) void lstm_step_kernel(
    int t,
    const float* __restrict__ g,      // [T, R, 64]
    const float* __restrict__ w_ih,   // [256, 64]
    const float* __restrict__ w_hh,   // [256, 64]
    const float* __restrict__ b_ih,   // [256]
    const float* __restrict__ b_hh,   // [256]
    const float* __restrict__ Hin,    // [R, 64]
    float* __restrict__ Hout,         // [R, 64]
    float* __restrict__ Cst,          // [R, 64] (in-place)
    const float* __restrict__ fc_w,   // [1, 64]
    const float* __restrict__ fc_b,   // [1]
    float* __restrict__ out)          // [R, T]
{
  __shared__ float py[16 * HIDD];

  const int tid  = threadIdx.x;
  const int wave = tid >> 5;
  const int lane = tid & 31;
  const int l16  = lane & 15;
  const int half = lane >> 4;
  const int m0 = blockIdx.x * 16;     // row tile
  const int n0 = wave * 16;           // hidden-col tile
  const int nh = n0 + l16;            // this lane's hidden column (B-operand N)

  v8f acc0 = {0.f,0.f,0.f,0.f,0.f,0.f,0.f,0.f};
  v8f acc1 = acc0, acc2 = acc0, acc3 = acc0;

  const float* gRow = g + ((size_t)t * RTOT + m0 + l16) * HIDD + half * 2;
  const float* hRow = Hin + (size_t)(m0 + l16) * HIDD + half * 2;
  const float* wi0 = w_ih + (size_t)(nh)       * HIDD + half * 2;
  const float* wi1 = w_ih + (size_t)(64 + nh)  * HIDD + half * 2;
  const float* wi2 = w_ih + (size_t)(128 + nh) * HIDD + half * 2;
  const float* wi3 = w_ih + (size_t)(192 + nh) * HIDD + half * 2;
  const float* wh0 = w_hh + (size_t)(nh)       * HIDD + half * 2;
  const float* wh1 = w_hh + (size_t)(64 + nh)  * HIDD + half * 2;
  const float* wh2 = w_hh + (size_t)(128 + nh) * HIDD + half * 2;
  const float* wh3 = w_hh + (size_t)(192 + nh) * HIDD + half * 2;

  #pragma unroll 4
  for (int kb = 0; kb < HIDD; kb += 4) {           // x_t @ w_ih^T
    v2f a = *(const v2f*)(gRow + kb);
    acc0 = wmma_f32_k4(a, *(const v2f*)(wi0 + kb), acc0);
    acc1 = wmma_f32_k4(a, *(const v2f*)(wi1 + kb), acc1);
    acc2 = wmma_f32_k4(a, *(const v2f*)(wi2 + kb), acc2);
    acc3 = wmma_f32_k4(a, *(const v2f*)(wi3 + kb), acc3);
  }
  #pragma unroll 4
  for (int kb = 0; kb < HIDD; kb += 4) {           // h_prev @ w_hh^T
    v2f a = *(const v2f*)(hRow + kb);
    acc0 = wmma_f32_k4(a, *(const v2f*)(wh0 + kb), acc0);
    acc1 = wmma_f32_k4(a, *(const v2f*)(wh1 + kb), acc1);
    acc2 = wmma_f32_k4(a, *(const v2f*)(wh2 + kb), acc2);
    acc3 = wmma_f32_k4(a, *(const v2f*)(wh3 + kb), acc3);
  }

  const float bi = b_ih[nh]        + b_hh[nh];
  const float bf = b_ih[64 + nh]   + b_hh[64 + nh];
  const float bg = b_ih[128 + nh]  + b_hh[128 + nh];
  const float bo = b_ih[192 + nh]  + b_hh[192 + nh];
  const float fw = fc_w[nh];

  #pragma unroll
  for (int r = 0; r < 8; ++r) {
    const int    mrow = m0 + r + half * 8;
    const size_t off  = (size_t)mrow * HIDD + nh;
    const float ig = sigm(acc0[r] + bi);
    const float fg = sigm(acc1[r] + bf);
    const float gg = tanhf(acc2[r] + bg);
    const float og = sigm(acc3[r] + bo);
    const float c  = fg * Cst[off] + ig * gg;
    const float h  = og * tanhf(c);
    Cst[off]  = c;
    Hout[off] = h;
    py[(r + half * 8) * HIDD + nh] = h * fw;       // partial of hs @ fc_w^T
  }
  __syncthreads();
  if (tid < 16) {                                  // deterministic row reduce
    float s = fc_b[0];
    const float* row = py + tid * HIDD;
    #pragma unroll
    for (int j = 0; j < HIDD; ++j) s += row[j];
    out[(size_t)(m0 + tid) * TT + t] = s;          // [B,N,T,1] = [R,T]
  }
}

// ---------------------------------------------------------------------------
extern "C" void kernel_launch(void* const* d_in, const int* in_sizes, int n_in,
                              void* d_out, int out_size, void* d_ws, size_t ws_size,
                              hipStream_t stream) {
  (void)in_sizes; (void)n_in; (void)out_size; (void)ws_size;
  const float* x    = (const float*)d_in[0];
  const float* A    = (const float*)d_in[1];
  const float* Ww   = (const float*)d_in[2];
  const float* Wb   = (const float*)d_in[3];
  const float* wih  = (const float*)d_in[4];
  const float* whh  = (const float*)d_in[5];
  const float* bih  = (const float*)d_in[6];
  const float* bhh  = (const float*)d_in[7];
  const float* fcw  = (const float*)d_in[8];
  const float* fcb  = (const float*)d_in[9];
  float* out = (float*)d_out;

  // Workspace carve-out (all offsets 256B aligned): ~403 MB total.
  char* p = (char*)d_ws;
  float* agg = (float*)p; p += (size_t)BB * NVTX * NCOL * sizeof(float);   // 44 MB
  float* g   = (float*)p; p += (size_t)TT * RTOT * HIDD * sizeof(float);   // 352 MB
  float* H0  = (float*)p; p += (size_t)RTOT * HIDD * sizeof(float);        // 2 MB
  float* H1  = (float*)p; p += (size_t)RTOT * HIDD * sizeof(float);        // 2 MB
  float* C   = (float*)p; p += (size_t)RTOT * HIDD * sizeof(float);        // 2 MB

  hipMemsetAsync(H0, 0, (size_t)RTOT * HIDD * sizeof(float), stream);
  hipMemsetAsync(C,  0, (size_t)RTOT * HIDD * sizeof(float), stream);

  // Stage 1: lagged aggregation (45 GFLOP on the fp32 WMMA pipe)
  dim3 gAgg(NCOL / 16, NVTX / 128, BB);   // 84 x 4 x 16 blocks, 4 waves each
  agg_wmma_kernel<<<gAgg, 128, 0, stream>>>(x, A, agg);

  // Stage 2: projection + exact GELU
  const size_t nproj = (size_t)RTOT * TT * HIDD;
  proj_gelu_kernel<<<(unsigned)((nproj + 255) / 256), 256, 0, stream>>>(agg, Ww, Wb, g);

  // Stage 3: sequential LSTM, one capture-friendly launch per timestep,
  // H ping-ponged (block reads Hin rows as WMMA-A, writes Hout; no races).
  for (int t = 0; t < TT; ++t) {
    const float* Hin = (t & 1) ? H1 : H0;
    float*       Hout = (t & 1) ? H0 : H1;
    lstm_step_kernel<<<RTOT / 16, 128, 0, stream>>>(
        t, g, wih, whh, bih, bhh, Hin, Hout, C, fcw, fcb, out);
  }
}